// SetExpansion_42975442764365
// MI455X (gfx1250) — compile-verified
//
#include <hip/hip_runtime.h>
#include <math.h>

// ---------------------------------------------------------------------------
// SetExpansion on MI455X (gfx1250): fp32 WMMA GEMMs + deterministic rank-select
// ---------------------------------------------------------------------------
#define CH 2048
#define AMAX 16384
#define KCH 32          // K-split chunks for mat-vec (64 rows each)

typedef __attribute__((ext_vector_type(2))) float v2f;
typedef __attribute__((ext_vector_type(8))) float v8f;

__device__ __forceinline__ float lrelu_f(float v) { return v >= 0.0f ? v : 0.2f * v; }

// ----------------------------- mat-vec (K-split) ---------------------------
// part[kc][j] = sum_{k in chunk kc} x[k] * W[k][j]
__global__ __launch_bounds__(256) void matvec_part_k(
    const float* __restrict__ xin, const float* __restrict__ W,
    float* __restrict__ part)
{
  const int j  = (blockIdx.x * 256 + threadIdx.x) * 4;   // gridDim.x == 2
  const int k0 = blockIdx.y * (CH / KCH);                // gridDim.y == KCH
  float4 acc = make_float4(0.f, 0.f, 0.f, 0.f);
  #pragma unroll 4
  for (int k = k0; k < k0 + (CH / KCH); ++k) {
    const float xv = xin[k];
    const float4 w = *reinterpret_cast<const float4*>(W + (size_t)k * CH + j);
    acc.x = fmaf(xv, w.x, acc.x);
    acc.y = fmaf(xv, w.y, acc.y);
    acc.z = fmaf(xv, w.z, acc.z);
    acc.w = fmaf(xv, w.w, acc.w);
  }
  *reinterpret_cast<float4*>(part + (size_t)blockIdx.y * CH + j) = acc;
}

__global__ __launch_bounds__(256) void matvec_red_k(
    const float* __restrict__ part, const float* __restrict__ bias,
    float* __restrict__ y, int act)
{
  const int j = blockIdx.x * 256 + threadIdx.x;          // gridDim.x == 8
  float s = bias[j];
  #pragma unroll 4
  for (int p = 0; p < KCH; ++p) s += part[(size_t)p * CH + j];
  y[j] = act ? lrelu_f(s) : s;
}

// ----------------------- length head: dot + clip --------------------------
__global__ __launch_bounds__(256) void finalize_l_k(
    const float* __restrict__ h, const float* __restrict__ lWo,
    const float* __restrict__ lbo, float* __restrict__ out_l)
{
  __shared__ float red[256];
  float s = 0.f;
  for (int i = threadIdx.x; i < CH; i += 256) s = fmaf(h[i], lWo[i], s);
  red[threadIdx.x] = s;
  __syncthreads();
  for (int off = 128; off > 0; off >>= 1) {
    if (threadIdx.x < off) red[threadIdx.x] += red[threadIdx.x + off];
    __syncthreads();
  }
  if (threadIdx.x == 0) {
    float l = red[0] + lbo[0];
    l = fminf(fabsf(l), 0.9999f);
    *out_l = l;
  }
}

// -------------------------- scores = members @ q ---------------------------
__global__ __launch_bounds__(256) void scores_k(
    const float* __restrict__ members, const float* __restrict__ q,
    float* __restrict__ sc)
{
  const int wave = threadIdx.x >> 5;
  const int lane = threadIdx.x & 31;
  const int row  = blockIdx.x * 8 + wave;                // gridDim.x == 2048
  const float* mp = members + (size_t)row * CH;
  float s = 0.f;
  #pragma unroll 4
  for (int i = lane; i < CH; i += 32) s = fmaf(mp[i], q[i], s);
  for (int off = 16; off > 0; off >>= 1) s += __shfl_down(s, off, 32);
  if (lane == 0) sc[row] = s;
}

// ------------- exact stable ascending rank -> top-c selection -------------
__global__ __launch_bounds__(256) void rank_select_k(
    const float* __restrict__ sc, int* __restrict__ idxsel, int c)
{
  __shared__ float tile[4096];
  const int e = blockIdx.x * 256 + threadIdx.x;          // gridDim.x == 64
  const float se = sc[e];
  int rank = 0;
  for (int base = 0; base < AMAX; base += 4096) {
    __syncthreads();
    for (int i = threadIdx.x; i < 4096; i += 256) tile[i] = sc[base + i];
    __syncthreads();
    for (int j = 0; j < 4096; ++j) {
      const float sj = tile[j];
      const int gj = base + j;
      rank += (sj < se) || (sj == se && gj < e);   // stable: index tiebreak
    }
  }
  const int thresh = AMAX - c;
  if (rank >= thresh) idxsel[rank - thresh] = e;   // ascending-score order
}

// ----------------------- positional encoding table ------------------------
__global__ __launch_bounds__(256) void posenc_k(float* __restrict__ p1, int cpad)
{
  const int r = blockIdx.x * 256 + threadIdx.x;
  if (r >= cpad) return;
  const float pos = 0.9999f * (float)r / 16383.0f;
  float* o = p1 + (size_t)r * 17;
  o[0] = pos;
  #pragma unroll
  for (int k = 0; k < 8; ++k) {
    const float ang = pos * (3.14159265358979f * (float)(1 << k));
    o[1 + k] = sinf(ang);
    o[9 + k] = cosf(ang);
  }
}

// ------------------------- WMMA f32 GEMM (16x16x4) ------------------------
// MODE 0: A = members[idx[r]], epilogue adds vec1[n] + p1[r]·Wp[:,n], lrelu
// MODE 1: A = h_prev, epilogue adds bias[n], lrelu
// MODE 2: A = h_prev, epilogue adds bias[n], no act
template <int MODE>
__global__ __launch_bounds__(256) void gemm_wmma_k(
    const float* __restrict__ Asrc, const int* __restrict__ idx,
    const float* __restrict__ B,    const float* __restrict__ bias,
    const float* __restrict__ Wp,   const float* __restrict__ p1tab,
    float* __restrict__ Cout, int Mrows)
{
  __shared__ float As[16 * 68];    // 16 rows x 64 K, stride 68 (bank-safe b64)
  __shared__ float Bs[64 * 136];   // 64 K x 128 N, stride 136 (disjoint halves)

  const int tid  = threadIdx.x;
  const int wave = tid >> 5;
  const int lane = tid & 31;
  const int lo   = lane & 15;
  const int hi   = lane >> 4;

  const int nblk = blockIdx.x * 128;       // gridDim.x == 16
  const int r0   = blockIdx.y * 16;        // gridDim.y == cpad/16
  const int ncol = nblk + wave * 16 + lo;  // this lane's N column

  // A staging map: thread -> (row 0..15, 4 consecutive K)
  const int arow = tid >> 4;
  const int acol = (tid & 15) * 4;
  const float* arowp;
  if (MODE == 0) {
    int g = r0 + arow;
    if (g >= Mrows) g = Mrows - 1;         // clamp tail rows (never stored)
    arowp = Asrc + (size_t)idx[g] * CH;    // fused gather of members[idx]
  } else {
    arowp = Asrc + (size_t)(r0 + arow) * CH;
  }

  v8f acc = {0.f, 0.f, 0.f, 0.f, 0.f, 0.f, 0.f, 0.f};

  for (int kb = 0; kb < CH; kb += 64) {
    // stage A tile (1 float4 / thread, coalesced 256B per row)
    const float4 av = *reinterpret_cast<const float4*>(arowp + kb + acol);
    As[arow * 68 + acol + 0] = av.x;
    As[arow * 68 + acol + 1] = av.y;
    As[arow * 68 + acol + 2] = av.z;
    As[arow * 68 + acol + 3] = av.w;
    // stage B tile (8 float4 / thread, coalesced 512B per row)
    #pragma unroll
    for (int i = 0; i < 8; ++i) {
      const int f    = tid + i * 256;
      const int brow = f >> 5;
      const int bcol = (f & 31) * 4;
      const float4 bv = *reinterpret_cast<const float4*>(
          B + (size_t)(kb + brow) * CH + nblk + bcol);
      Bs[brow * 136 + bcol + 0] = bv.x;
      Bs[brow * 136 + bcol + 1] = bv.y;
      Bs[brow * 136 + bcol + 2] = bv.z;
      Bs[brow * 136 + bcol + 3] = bv.w;
    }
    __syncthreads();

    // 16 x V_WMMA_F32_16X16X4_F32 per 64-K chunk
    #pragma unroll
    for (int kk = 0; kk < 64; kk += 4) {
      // A frag: lane(0-15)=row M, K pair {k,k+1}; lanes 16-31: K pair {k+2,k+3}
      const v2f a = *reinterpret_cast<const v2f*>(&As[lo * 68 + kk + 2 * hi]);
      // B frag: v0 = B[k + 2*hi][n], v1 = B[k + 1 + 2*hi][n]
      v2f b;
      b.x = Bs[(kk + 2 * hi) * 136 + wave * 16 + lo];
      b.y = Bs[(kk + 2 * hi + 1) * 136 + wave * 16 + lo];
      acc = __builtin_amdgcn_wmma_f32_16x16x4_f32(
          false, a, false, b, (short)0, acc, false, false);
    }
    __syncthreads();
  }

  // epilogue: C layout -> element v of lane: M = v + 8*hi, N = lo
  if (MODE == 0) {
    float wp[17];
    #pragma unroll
    for (int t = 0; t < 17; ++t) wp[t] = Wp[(size_t)t * CH + ncol];
    const float basev = bias[ncol];        // vec1 (x-part + rbi)
    #pragma unroll
    for (int v = 0; v < 8; ++v) {
      const int row = r0 + v + 8 * hi;
      if (row < Mrows) {
        const float* pp = p1tab + (size_t)row * 17;
        float add = basev;
        #pragma unroll
        for (int t = 0; t < 17; ++t) add = fmaf(pp[t], wp[t], add);
        Cout[(size_t)row * CH + ncol] = lrelu_f(acc[v] + add);
      }
    }
  } else {
    const float bv = bias[ncol];
    #pragma unroll
    for (int v = 0; v < 8; ++v) {
      const int row = r0 + v + 8 * hi;
      if (row < Mrows) {
        float val = acc[v] + bv;
        if (MODE == 1) val = lrelu_f(val);
        Cout[(size_t)row * CH + ncol] = val;
      }
    }
  }
}

// ---------------------------------------------------------------------------
extern "C" void kernel_launch(void* const* d_in, const int* in_sizes, int n_in,
                              void* d_out, int out_size, void* d_ws, size_t ws_size,
                              hipStream_t stream) {
  const float* x       = (const float*)d_in[0];
  const float* members = (const float*)d_in[1];
  const float* qW      = (const float*)d_in[2];
  const float* qb      = (const float*)d_in[3];
  const float* lW      = (const float*)d_in[4];
  const float* lb      = (const float*)d_in[5];
  const float* lWo     = (const float*)d_in[6];
  const float* lbo     = (const float*)d_in[7];
  const float* rWi     = (const float*)d_in[8];
  const float* rbi     = (const float*)d_in[9];
  const float* rWh     = (const float*)d_in[10];
  const float* rbh     = (const float*)d_in[11];
  const float* rWo     = (const float*)d_in[12];
  const float* rbo     = (const float*)d_in[13];
  float* out = (float*)d_out;

  // out = [c,2048] flattened followed by l[1,1]  ->  c is recoverable.
  int c = (out_size - 1) / CH;
  if (c < 1) c = 1;
  const int cpad = (c + 15) & ~15;
  const size_t W2 = (size_t)CH * CH;

  // workspace layout (floats)
  float* w      = (float*)d_ws;
  float* hlA    = w;                    // 2048
  float* hlB    = hlA + CH;             // 2048
  float* hqA    = hlB + CH;             // 2048
  float* hqB    = hqA + CH;             // 2048
  float* qv     = hqB + CH;             // 2048
  float* vec1   = qv + CH;              // 2048
  float* mvpart = vec1 + CH;            // KCH*2048
  float* sc     = mvpart + (size_t)KCH * CH;  // 16384
  int*   idxsel = (int*)(sc + AMAX);          // 16384 ints
  float* p1     = (float*)(idxsel + AMAX);    // 16384*20 reserved
  float* h1     = p1 + (size_t)AMAX * 20;     // cpad*2048
  float* h2     = h1 + (size_t)cpad * CH;     // cpad*2048

  const dim3 b256(256);
  const dim3 gMV(2, KCH);
  auto mv = [&](const float* xin, const float* Wm, const float* bm,
                float* yv, int act) {
    matvec_part_k<<<gMV, b256, 0, stream>>>(xin, Wm, mvpart);
    matvec_red_k<<<dim3(8), b256, 0, stream>>>(mvpart, bm, yv, act);
  };

  // length MLP chain -> l (written to out[c*2048])
  mv(x,   lW + 0 * W2, lb + 0 * CH, hlA, 1);
  mv(hlA, lW + 1 * W2, lb + 1 * CH, hlB, 1);
  mv(hlB, lW + 2 * W2, lb + 2 * CH, hlA, 1);
  finalize_l_k<<<dim3(1), b256, 0, stream>>>(hlA, lWo, lbo, out + (size_t)c * CH);

  // query MLP chain -> q
  mv(x,   qW + 0 * W2, qb + 0 * CH, hqA, 1);
  mv(hqA, qW + 1 * W2, qb + 1 * CH, hqB, 1);
  mv(hqB, qW + 2 * W2, qb + 2 * CH, hqA, 1);
  mv(hqA, qW + 3 * W2, qb + 3 * CH, qv, 0);

  // vec1 = x @ rWi[2048:4096] + rbi  (broadcast term of first reduce layer)
  mv(x, rWi + W2, rbi, vec1, 0);

  // scores, top-c selection (ascending), positional table
  scores_k<<<dim3(AMAX / 8), b256, 0, stream>>>(members, qv, sc);
  rank_select_k<<<dim3(AMAX / 256), b256, 0, stream>>>(sc, idxsel, c);
  posenc_k<<<dim3((cpad + 255) / 256), b256, 0, stream>>>(p1, cpad);

  // reduce stack: 4 WMMA GEMMs
  const dim3 gG(16, cpad / 16);
  gemm_wmma_k<0><<<gG, b256, 0, stream>>>(members, idxsel, rWi, vec1,
                                          rWi + 2 * W2, p1, h1, c);
  gemm_wmma_k<1><<<gG, b256, 0, stream>>>(h1, nullptr, rWh + 0 * W2,
                                          rbh + 0 * CH, nullptr, nullptr, h2, c);
  gemm_wmma_k<1><<<gG, b256, 0, stream>>>(h2, nullptr, rWh + 1 * W2,
                                          rbh + 1 * CH, nullptr, nullptr, h1, c);
  gemm_wmma_k<2><<<gG, b256, 0, stream>>>(h1, nullptr, rWo,
                                          rbo, nullptr, nullptr, out, c);

  (void)in_sizes; (void)n_in; (void)ws_size;
}